// InvertibleConv1x1_29583734735107
// MI455X (gfx1250) — compile-verified
//
#include <hip/hip_runtime.h>
#include <math.h>

typedef __attribute__((ext_vector_type(2))) float v2f;
typedef __attribute__((ext_vector_type(8))) float v8f;

#define NCH   96        // channels C
#define NHW   16384     // H*W
#define NB    16        // batch
#define LDSW  98        // padded LDS stride for W (98 mod 64 = 34 -> conflict-free b64)
#define NT    4         // N subtiles (16 cols each) per wave
#define MTT   6         // M tiles = 96/16
#define KSTEPS 24       // 96/4
#define WPB   8         // waves per block (256 threads)

// ---------------- Phase 1: build w = (prod of Householders) @ (triu(U,1)+diag(S)),
// and logdet_out. One block. Each thread owns one row of q; rows are independent
// across Householder steps, so no barriers needed inside the scan.
__global__ __launch_bounds__(128)
void build_w_kernel(const float* __restrict__ S, const float* __restrict__ U,
                    const float* __restrict__ v, const float* __restrict__ logdet_in,
                    float* __restrict__ w_out, float* __restrict__ logdet_out)
{
    __shared__ float q[NCH][NCH + 1];   // stride 97 (odd) -> conflict-free b32
    __shared__ float vv[NCH][NCH];      // broadcast reads, no pad needed
    const int t = threadIdx.x;

    for (int idx = t; idx < NCH * NCH; idx += blockDim.x) {
        int r = idx / NCH, c = idx % NCH;
        vv[r][c] = v[idx];
        q[r][c]  = (r == c) ? 1.0f : 0.0f;
    }
    __syncthreads();

    if (t < NCH) {
        // sequential scan over 96 reflections; this thread updates row t of q
        for (int i = 0; i < NCH; ++i) {
            float vtv = 0.0f, u = 0.0f;
            for (int c = 0; c < NCH; ++c) {
                float x = vv[i][c];
                vtv += x * x;
                u   += q[t][c] * x;
            }
            float s = 2.0f * u / vtv;
            for (int c = 0; c < NCH; ++c)
                q[t][c] -= s * vv[i][c];
        }
        // w[t][c] = q[t][c]*S[c] + sum_{k<c} q[t][k]*U[k][c]
        for (int c = 0; c < NCH; ++c) {
            float acc = q[t][c] * S[c];
            for (int k = 0; k < c; ++k)
                acc += q[t][k] * U[k * NCH + c];
            w_out[t * NCH + c] = acc;
        }
    }
    if (t < NB) {
        float s = 0.0f;
        for (int c = 0; c < NCH; ++c)
            s += logf(fabsf(S[c]));
        logdet_out[t] = logdet_in[t] + s * (float)NHW;
    }
}

// ---------------- Phase 2: out[b] = w @ z[b]   (96x96) x (96x16384), fp32 WMMA.
// Each wave: one (b, 64-column) stripe; 6 M-tiles x 4 N-subtiles accumulators.
// z and out are touch-once streams (~200 MB vs 192 MB L2) -> non-temporal hints.
__global__ __launch_bounds__(256)
void conv1x1_wmma_kernel(const float* __restrict__ z, const float* __restrict__ w,
                         float* __restrict__ out)
{
    __shared__ float Wl[NCH][LDSW];
    const int tid = threadIdx.x;

    for (int idx = tid; idx < NCH * NCH; idx += 256)
        Wl[idx / NCH][idx % NCH] = w[idx];
    __syncthreads();

    const int lane  = tid & 31;
    const int wid   = tid >> 5;
    const int lmod  = lane & 15;
    const int lhalf = lane >> 4;           // 0: K+0/1, N 0..15   1: K+2/3

    const int ntilesPerB = NHW / (16 * NT);            // 256
    const int task = blockIdx.x * WPB + wid;           // 0..4095 exactly
    const int b    = task / ntilesPerB;
    const int n0   = (task % ntilesPerB) * (16 * NT);

    const float* __restrict__ zb = z   + (size_t)b * NCH * NHW;
    float* __restrict__       ob = out + (size_t)b * NCH * NHW;

    v8f acc[MTT][NT];
#pragma unroll
    for (int m = 0; m < MTT; ++m)
#pragma unroll
        for (int n = 0; n < NT; ++n)
            acc[m][n] = (v8f){0.f, 0.f, 0.f, 0.f, 0.f, 0.f, 0.f, 0.f};

    for (int kk = 0; kk < KSTEPS; ++kk) {
        const int k = kk * 4 + 2 * lhalf;
        // A fragments: lane -> row M = 16*m + lmod, regs -> K pair (LDS b64)
        v2f a[MTT];
#pragma unroll
        for (int m = 0; m < MTT; ++m) {
            const float* p = &Wl[m * 16 + lmod][k];
            a[m][0] = p[0];
            a[m][1] = p[1];
        }
        // B fragments: lane -> col N = n0 + 16*n + lmod, regs -> K pair
        // (global, streamed once -> non-temporal)
        v2f bf[NT];
#pragma unroll
        for (int n = 0; n < NT; ++n) {
            const float* p = zb + (size_t)k * NHW + (n0 + n * 16 + lmod);
            bf[n][0] = __builtin_nontemporal_load(p);
            bf[n][1] = __builtin_nontemporal_load(p + NHW);
        }
#pragma unroll
        for (int m = 0; m < MTT; ++m)
#pragma unroll
            for (int n = 0; n < NT; ++n)
                acc[m][n] = __builtin_amdgcn_wmma_f32_16x16x4_f32(
                    false, a[m], false, bf[n], (short)0, acc[m][n], false, false);
    }

    // D layout: VGPR j -> row M = 16*m + j + 8*lhalf, lane%16 -> col
#pragma unroll
    for (int m = 0; m < MTT; ++m)
#pragma unroll
        for (int n = 0; n < NT; ++n)
#pragma unroll
            for (int j = 0; j < 8; ++j) {
                int row = m * 16 + j + 8 * lhalf;
                __builtin_nontemporal_store(acc[m][n][j],
                    ob + (size_t)row * NHW + (n0 + n * 16 + lmod));
            }
}

extern "C" void kernel_launch(void* const* d_in, const int* in_sizes, int n_in,
                              void* d_out, int out_size, void* d_ws, size_t ws_size,
                              hipStream_t stream)
{
    const float* z      = (const float*)d_in[0];
    const float* logdet = (const float*)d_in[1];
    const float* S      = (const float*)d_in[2];
    const float* U      = (const float*)d_in[3];
    const float* v      = (const float*)d_in[4];

    float* zout       = (float*)d_out;
    float* logdet_out = zout + (size_t)NB * NCH * NHW;
    float* wbuf       = (float*)d_ws;   // 96*96 floats = 36 KB scratch

    build_w_kernel<<<1, 128, 0, stream>>>(S, U, v, logdet, wbuf, logdet_out);

    const int tasks  = NB * (NHW / (16 * NT));   // 4096 waves
    const int blocks = tasks / WPB;              // 512 blocks of 256 threads
    conv1x1_wmma_kernel<<<blocks, 256, 0, stream>>>(z, wbuf, zout);
}